// affine_layer_22565758173448
// MI455X (gfx1250) — compile-verified
//
#include <hip/hip_runtime.h>
#include <math.h>

// Problem dims (fixed by the reference).
#define C_  32
#define D_  64
#define H_  192
#define W_  160
#define HW_  (H_ * W_)
#define DHW_ (D_ * H_ * W_)

typedef __attribute__((ext_vector_type(2))) float v2f;
typedef __attribute__((ext_vector_type(8))) float v8f;

__device__ __forceinline__ int iclamp(int v, int lo, int hi) {
    return v < lo ? lo : (v > hi ? hi : v);
}

// One block per (w, h). Block = 256 threads = 8 wave32 waves.
// Every wave computes the affine-grid coordinates for d = 0..63 with
// V_WMMA_F32_16X16X4_F32 (EXEC is guaranteed all-1s since there is no
// branch), wave 0 scatters them to LDS; then every thread does trilinear
// gathers for its (d, channel-group) and writes the contiguous [C, D]
// output chunk.
__global__ __launch_bounds__(256)
void affine_grid_sample_3d_kernel(const float* __restrict__ inp,
                                  const float* __restrict__ theta,
                                  float* __restrict__ out) {
    __shared__ float sx[D_];
    __shared__ float sy[D_];
    __shared__ float sz[D_];

    const int w    = blockIdx.x;
    const int h    = blockIdx.y;
    const int tid  = threadIdx.x;
    const int lane = tid & 31;
    const bool w0  = tid < 32;

    // ---- Affine grid via WMMA: D[m][n] = sum_k A[m][k] * B[k][n] ----
    // A (16x4 f32): lane l<16 -> row m=l,    VGPR0=K0, VGPR1=K1
    //               lane l>=16 -> row m=l-16, VGPR0=K2, VGPR1=K3
    // B (4x16 f32): lane l -> col n=l%16; VGPR0 = K0 (lo half) / K2 (hi),
    //               VGPR1 = K1 (lo) / K3 (hi).  B[k][n] = theta[n*4+k].
    const int  m  = lane & 15;
    const bool hi = lane >= 16;
    const int  n  = m;

    v2f b;
    if (n < 3) {
        b.x = theta[n * 4 + (hi ? 2 : 0)];
        b.y = theta[n * 4 + (hi ? 3 : 1)];
    } else {
        b.x = 0.0f;
        b.y = 0.0f;
    }

    // Compile-time reciprocals: avoid IEEE divide sequences in codegen.
    const float gx = fmaf((float)w, 2.0f / (float)(W_ - 1), -1.0f);
    const float gy = fmaf((float)h, 2.0f / (float)(H_ - 1), -1.0f);

    #pragma unroll
    for (int dbase = 0; dbase < D_; dbase += 16) {
        const float gz = fmaf((float)(dbase + m), 2.0f / (float)(D_ - 1), -1.0f);
        v2f a;
        if (!hi) { a.x = gx; a.y = gy;   }   // K0 = gx, K1 = gy
        else     { a.x = gz; a.y = 1.0f; }   // K2 = gz, K3 = 1
        v8f c = {};
        c = __builtin_amdgcn_wmma_f32_16x16x4_f32(
                /*neg_a=*/false, a, /*neg_b=*/false, b,
                /*c_mod=*/(short)0, c, /*reuse_a=*/false, /*reuse_b=*/false);
        // D fragment: lane holds D[r + 8*hi][n] in c[r], r = 0..7.
        if (w0 && n < 3) {
            float* dst = (n == 0) ? sx : (n == 1) ? sy : sz;
            const int mb = dbase + (hi ? 8 : 0);
            #pragma unroll
            for (int r = 0; r < 8; ++r) dst[mb + r] = c[r];
        }
    }
    __syncthreads();

    // ---- Trilinear sample (zeros padding, align_corners=True) ----
    const int d  = tid & 63;   // output D index (stride-1 in output)
    const int cg = tid >> 6;   // channel subgroup 0..3

    const float fx = (sx[d] + 1.0f) * (0.5f * (float)(W_ - 1));
    const float fy = (sy[d] + 1.0f) * (0.5f * (float)(H_ - 1));
    const float fz = (sz[d] + 1.0f) * (0.5f * (float)(D_ - 1));

    const float flx = floorf(fx), fly = floorf(fy), flz = floorf(fz);
    const int ix0 = (int)flx, iy0 = (int)fly, iz0 = (int)flz;
    const int ix1 = ix0 + 1,  iy1 = iy0 + 1,  iz1 = iz0 + 1;

    // Per-axis lerp weights with validity folded in (separable zero padding).
    float u1 = fx - flx, u0 = 1.0f - u1;
    float v1 = fy - fly, v0 = 1.0f - v1;
    float t1 = fz - flz, t0 = 1.0f - t1;
    u0 *= (ix0 >= 0 && ix0 < W_) ? 1.0f : 0.0f;
    u1 *= (ix1 >= 0 && ix1 < W_) ? 1.0f : 0.0f;
    v0 *= (iy0 >= 0 && iy0 < H_) ? 1.0f : 0.0f;
    v1 *= (iy1 >= 0 && iy1 < H_) ? 1.0f : 0.0f;
    t0 *= (iz0 >= 0 && iz0 < D_) ? 1.0f : 0.0f;
    t1 *= (iz1 >= 0 && iz1 < D_) ? 1.0f : 0.0f;

    const int cx0 = iclamp(ix0, 0, W_ - 1), cx1 = iclamp(ix1, 0, W_ - 1);
    const int cy0 = iclamp(iy0, 0, H_ - 1), cy1 = iclamp(iy1, 0, H_ - 1);
    const int cz0 = iclamp(iz0, 0, D_ - 1), cz1 = iclamp(iz1, 0, D_ - 1);

    // Channel-invariant corner offsets.
    const int oz0 = cz0 * HW_, oz1 = cz1 * HW_;
    const int oy0 = cy0 * W_,  oy1 = cy1 * W_;
    const int o000 = oz0 + oy0 + cx0, o001 = oz0 + oy0 + cx1;
    const int o010 = oz0 + oy1 + cx0, o011 = oz0 + oy1 + cx1;
    const int o100 = oz1 + oy0 + cx0, o101 = oz1 + oy0 + cx1;
    const int o110 = oz1 + oy1 + cx0, o111 = oz1 + oy1 + cx1;

    // Output chunk for this (w,h): contiguous C_*D_ floats.
    float* ob = out + ((size_t)(w * H_ + h) * C_) * D_ + d;

    #pragma unroll
    for (int i = 0; i < 8; ++i) {
        const int c = cg + 4 * i;                 // 0..31
        const float* pc = inp + (size_t)c * DHW_;
        const float acc =
            t0 * (v0 * (u0 * pc[o000] + u1 * pc[o001]) +
                  v1 * (u0 * pc[o010] + u1 * pc[o011])) +
            t1 * (v0 * (u0 * pc[o100] + u1 * pc[o101]) +
                  v1 * (u0 * pc[o110] + u1 * pc[o111]));
        ob[c * D_] = acc;                         // coalesced along d
    }
}

extern "C" void kernel_launch(void* const* d_in, const int* in_sizes, int n_in,
                              void* d_out, int out_size, void* d_ws, size_t ws_size,
                              hipStream_t stream) {
    (void)in_sizes; (void)n_in; (void)d_ws; (void)ws_size; (void)out_size;
    const float* inp   = (const float*)d_in[0];   // [1,32,64,192,160] f32
    const float* theta = (const float*)d_in[1];   // [12] f32
    float* out = (float*)d_out;                   // [1,160,192,32,64] f32

    dim3 grid(W_, H_);   // 160 x 192 blocks, one per (w,h)
    dim3 block(256);     // 8 wave32 waves; covers (cg=4) x (d=64)
    affine_grid_sample_3d_kernel<<<grid, block, 0, stream>>>(inp, theta, out);
}